// DSPNSetAutoencoder_55628416418430
// MI455X (gfx1250) — compile-verified
//
#include <hip/hip_runtime.h>
#include <math.h>

// Problem constants (from reference)
#define DD     64      // element dim
#define HH     2048    // hidden dim
#define LL     1024    // latent dim
#define NSET   512     // set size (decode_count == MAXN)
#define BB     16      // batch
#define NSTEPS 10
#define GSTEP  0.45f

typedef float v8f __attribute__((ext_vector_type(8)));
typedef float v4f __attribute__((ext_vector_type(4)));
typedef float v2f __attribute__((ext_vector_type(2)));

// Packed W1 fragment tables in workspace (one-time build):
//  Bpack: forward/Q-recompute B fragments, (HH/16) n-tiles x 16 k-steps x 32
//         lanes, one float2 per lane -> one coalesced global_load_b64 / WMMA.
//  Tpack: backward (W1^T) B fragments, (HH/16) k-chunks x 4 k-steps x 4
//         d-tiles x 32 lanes, one float2 per lane.
#define BPACK_V2 ((HH / 16) * 16 * 32)         // 65536 float2
#define TPACK_V2 ((HH / 16) * 4 * 4 * 32)      // 65536 float2

static __device__ __forceinline__ v8f wmma_f32(v2f a, v2f b, v8f c) {
  // D = A(16x4,f32) * B(4x16,f32) + C(16x16,f32)   -> v_wmma_f32_16x16x4_f32
  return __builtin_amdgcn_wmma_f32_16x16x4_f32(false, a, false, b, (short)0, c,
                                               false, false);
}

// Column-sum of relu(M(512x64) @ W1(64x2048) + b1) accumulated into s_s[2048]
// via LDS float atomics.  s_s zeroed and barriers handled by caller.
// A fragment (16x4 f32): lane m = lane&15; VGPR0 holds K=0 (lanes 0-15) /
// K=2 (lanes 16-31), VGPR1 holds K=1 / K=3.
// B fragments come pre-packed from Bpack (1 b64 load per WMMA).
static __device__ void fwd_colsum(const float* __restrict__ M,
                                  const v2f* __restrict__ Bp,
                                  const float* __restrict__ s_b1,
                                  float* __restrict__ s_s,
                                  int wave, int lane) {
  const int m16 = lane & 15;
  const int grp = lane >> 4;
  for (int mt = wave; mt < NSET / 16; mt += 8) {
    // cache A fragments for this 16-row tile of M (K = 0..63)
    v2f a[16];
    const float* mrow = M + (mt * 16 + m16) * DD + 2 * grp;
#pragma unroll
    for (int i = 0; i < 16; ++i) a[i] = *(const v2f*)(mrow + 4 * i);

    // two n-tiles per iteration: two independent WMMA chains
    for (int nt = 0; nt < HH / 16; nt += 2) {
      v8f acc0 = {0.f, 0.f, 0.f, 0.f, 0.f, 0.f, 0.f, 0.f};
      v8f acc1 = {0.f, 0.f, 0.f, 0.f, 0.f, 0.f, 0.f, 0.f};
#pragma unroll
      for (int i = 0; i < 16; ++i) {
        const v2f b0 = Bp[((nt + 0) * 16 + i) * 32 + lane];
        const v2f b1 = Bp[((nt + 1) * 16 + i) * 32 + lane];
        acc0 = wmma_f32(a[i], b0, acc0);
        acc1 = wmma_f32(a[i], b1, acc1);
      }
      const int n0 = (nt + 0) * 16 + m16;
      const int n1 = (nt + 1) * 16 + m16;
      const float bias0 = s_b1[n0];
      const float bias1 = s_b1[n1];
      float p0 = 0.f, p1 = 0.f;
#pragma unroll
      for (int r = 0; r < 8; ++r) {
        const float h0 = acc0[r] + bias0;
        const float h1 = acc1[r] + bias1;
        p0 += h0 > 0.f ? h0 : 0.f;
        p1 += h1 > 0.f ? h1 : 0.f;
      }
      atomicAdd(&s_s[n0], p0);  // ds_add_f32
      atomicAdd(&s_s[n1], p1);
    }
  }
}

// ---------------------------------------------------------------------------
// One-time: build fragment-major W1 tables in workspace.
// ---------------------------------------------------------------------------
__global__ void __launch_bounds__(256)
pack_w1(const float* __restrict__ W1, float* __restrict__ Bpack,
        float* __restrict__ Tpack) {
  const int idx = blockIdx.x * 256 + threadIdx.x;  // over BPACK_V2 + TPACK_V2
  if (idx < BPACK_V2) {
    const int lane = idx & 31;
    const int t    = idx >> 5;        // nt*16 + i
    const int i    = t & 15;
    const int nt   = t >> 4;
    const int m16  = lane & 15;
    const int grp  = lane >> 4;
    const int n    = nt * 16 + m16;
    const int k    = 4 * i + 2 * grp;
    v2f val;
    val.x = W1[k * HH + n];
    val.y = W1[(k + 1) * HH + n];
    *(v2f*)(Bpack + 2 * (size_t)idx) = val;
  } else if (idx < BPACK_V2 + TPACK_V2) {
    const int j    = idx - BPACK_V2;
    const int lane = j & 31;
    const int t    = j >> 5;          // (kc*4 + i)*4 + dt
    const int dt   = t & 3;
    const int ti   = t >> 2;
    const int i    = ti & 3;
    const int kc   = ti >> 2;
    const int m16  = lane & 15;
    const int grp  = lane >> 4;
    const int col  = dt * 16 + m16;           // D index
    const int kk   = kc * 16 + 4 * i + 2 * grp;  // H index
    // (W1^T[kk][col], W1^T[kk+1][col]) == contiguous pair in W1 row col
    const v2f val = *(const v2f*)(W1 + col * HH + kk);
    *(v2f*)(Tpack + 2 * (size_t)j) = val;
  }
}

// ---------------------------------------------------------------------------
// Kernel 1: encoder.  One workgroup per batch element.
// ---------------------------------------------------------------------------
__global__ void __launch_bounds__(256)
encode_kernel(const float* __restrict__ x, const float* __restrict__ Bpack,
              const float* __restrict__ b1, const float* __restrict__ W2,
              const float* __restrict__ b2, const float* __restrict__ Ws1,
              const float* __restrict__ bs1, const float* __restrict__ Ws2,
              const float* __restrict__ bs2, float* __restrict__ latent,
              float* __restrict__ pred) {
  __shared__ float s_s[HH];
  __shared__ float s_b1[HH];
  __shared__ float s_z[LL];
  __shared__ float s_raw;

  const int tid  = threadIdx.x;
  const int b    = blockIdx.x;
  const int wave = tid >> 5;
  const int lane = tid & 31;

  for (int i = tid; i < HH; i += 256) { s_b1[i] = b1[i]; s_s[i] = 0.f; }
  if (tid == 0) s_raw = bs2[0];
  __syncthreads();

  fwd_colsum(x + (size_t)b * NSET * DD, (const v2f*)Bpack, s_b1, s_s, wave,
             lane);
  __syncthreads();

  // z = s @ W2 + N*b2
  for (int l = tid; l < LL; l += 256) {
    float z = (float)NSET * b2[l];
    for (int k = 0; k < HH; ++k) z += s_s[k] * W2[k * LL + l];
    s_z[l] = z;
    latent[b * LL + l] = z;
  }
  __syncthreads();

  // raw = relu(z @ Ws1 + bs1) . Ws2 + bs2
  float part = 0.f;
  for (int jj = 0; jj < 8; ++jj) {
    const int j = tid + 256 * jj;
    float t = bs1[j];
    for (int l = 0; l < LL; ++l) t += s_z[l] * Ws1[l * HH + j];
    t = t > 0.f ? t : 0.f;
    part += t * Ws2[j];
  }
  atomicAdd(&s_raw, part);
  __syncthreads();
  if (tid == 0) {
    const float sg = 1.f / (1.f + __expf(-s_raw));
    pred[b] = sg * (float)NSET;
  }
}

// ---------------------------------------------------------------------------
// Kernel 2: 10 gradient-descent steps, one workgroup per batch element.
// ---------------------------------------------------------------------------
__global__ void __launch_bounds__(256)
decode_kernel(const float* __restrict__ Bpack, const float* __restrict__ Tpack,
              const float* __restrict__ b1, const float* __restrict__ W2,
              const float* __restrict__ b2, const float* __restrict__ init_set,
              const float* __restrict__ latent, float* __restrict__ recon) {
  __shared__ float s_s[HH];
  __shared__ float s_u[HH];
  __shared__ float s_b1[HH];
  __shared__ float s_v[LL];
  __shared__ float s_zt[LL];
  __shared__ float s_wstage[8 * 256];  // per-wave 16x16 C->A relayout buffer

  const int tid  = threadIdx.x;
  const int b    = blockIdx.x;
  const int wave = tid >> 5;
  const int lane = tid & 31;
  const int m16  = lane & 15;
  const int grp  = lane >> 4;

  const v2f* Bp = (const v2f*)Bpack;
  const v2f* Tp = (const v2f*)Tpack;
  float* G = recon + (size_t)b * NSET * DD;

  for (int i = tid; i < HH; i += 256) s_b1[i] = b1[i];
  for (int i = tid; i < LL; i += 256) s_zt[i] = latent[b * LL + i];
  for (int i = tid; i < NSET * DD; i += 256) G[i] = init_set[i];
  __threadfence_block();
  __syncthreads();

  for (int step = 0; step < NSTEPS; ++step) {
    for (int i = tid; i < HH; i += 256) s_s[i] = 0.f;
    __syncthreads();

    fwd_colsum(G, Bp, s_b1, s_s, wave, lane);
    __syncthreads();

    // v = (2/L) * (s W2 + N*b2 - zt)
    for (int l = tid; l < LL; l += 256) {
      float z = (float)NSET * b2[l];
      for (int k = 0; k < HH; ++k) z += s_s[k] * W2[k * LL + l];
      s_v[l] = (2.0f / (float)LL) * (z - s_zt[l]);
    }
    __syncthreads();

    // u = W2 v   (rows of W2 contiguous -> float4 loads)
    for (int k = tid; k < HH; k += 256) {
      const v4f* w2r = (const v4f*)(W2 + k * LL);
      float uu = 0.f;
      for (int l = 0; l < LL / 4; ++l) {
        const v4f w = w2r[l];
        uu += w.x * s_v[4 * l + 0] + w.y * s_v[4 * l + 1] +
              w.z * s_v[4 * l + 2] + w.w * s_v[4 * l + 3];
      }
      s_u[k] = uu;
    }
    __syncthreads();

    // backward: each wave owns 4 of the 32 row-tiles of G
    float* wst = s_wstage + wave * 256;
    for (int mt = wave; mt < NSET / 16; mt += 8) {
      v2f a[16];
      const float* grow = G + (mt * 16 + m16) * DD + 2 * grp;
#pragma unroll
      for (int i = 0; i < 16; ++i) a[i] = *(const v2f*)(grow + 4 * i);

      v8f gacc[4];
#pragma unroll
      for (int dt = 0; dt < 4; ++dt)
        gacc[dt] = (v8f){0.f, 0.f, 0.f, 0.f, 0.f, 0.f, 0.f, 0.f};

      for (int kc = 0; kc < HH / 16; ++kc) {
        const int n = kc * 16 + m16;
        // recompute 16x16 tile of H = G W1 + b1 for this column chunk
        v8f q = {0.f, 0.f, 0.f, 0.f, 0.f, 0.f, 0.f, 0.f};
#pragma unroll
        for (int i = 0; i < 16; ++i) {
          const v2f bf = Bp[(kc * 16 + i) * 32 + lane];
          q = wmma_f32(a[i], bf, q);
        }
        // masked upstream value, staged C-layout -> A-layout through LDS
        const float bias = s_b1[n];
        const float uval = s_u[n];
#pragma unroll
        for (int r = 0; r < 8; ++r) {
          const float wv = (q[r] + bias) > 0.f ? uval : 0.f;
          wst[(grp * 8 + r) * 16 + m16] = wv;
        }
        asm volatile("s_wait_dscnt 0" ::: "memory");
        v2f wa[4];
#pragma unroll
        for (int i = 0; i < 4; ++i)
          wa[i] = *(const v2f*)(wst + m16 * 16 + 4 * i + 2 * grp);

        // grad_tile += Wmask(16x16) @ W1T(16x64) : 4 independent chains
#pragma unroll
        for (int i = 0; i < 4; ++i) {
#pragma unroll
          for (int dt = 0; dt < 4; ++dt) {
            const v2f bf = Tp[((kc * 4 + i) * 4 + dt) * 32 + lane];
            gacc[dt] = wmma_f32(wa[i], bf, gacc[dt]);
          }
        }
      }
      // G -= STEP * grad
#pragma unroll
      for (int dt = 0; dt < 4; ++dt) {
#pragma unroll
        for (int r = 0; r < 8; ++r) {
          const int row = mt * 16 + grp * 8 + r;
          const int col = dt * 16 + m16;
          G[row * DD + col] -= GSTEP * gacc[dt][r];
        }
      }
    }
    __threadfence_block();
    __syncthreads();
  }
}

extern "C" void kernel_launch(void* const* d_in, const int* in_sizes, int n_in,
                              void* d_out, int out_size, void* d_ws,
                              size_t ws_size, hipStream_t stream) {
  const float* x        = (const float*)d_in[0];
  const float* W1       = (const float*)d_in[1];
  const float* b1       = (const float*)d_in[2];
  const float* W2       = (const float*)d_in[3];
  const float* b2       = (const float*)d_in[4];
  const float* Ws1      = (const float*)d_in[5];
  const float* bs1      = (const float*)d_in[6];
  const float* Ws2      = (const float*)d_in[7];
  const float* bs2      = (const float*)d_in[8];
  const float* init_set = (const float*)d_in[9];
  // d_in[10] = decode_count == MAXN (compile-time NSET)

  float* out    = (float*)d_out;
  float* recon  = out;                                   // B*N*D
  float* latent = out + (size_t)BB * NSET * DD;          // B*L
  float* pred   = latent + (size_t)BB * LL;              // B

  float* Bpack = (float*)d_ws;                           // 512 KB
  float* Tpack = Bpack + 2 * (size_t)BPACK_V2;           // 512 KB

  pack_w1<<<(BPACK_V2 + TPACK_V2) / 256, 256, 0, stream>>>(W1, Bpack, Tpack);
  encode_kernel<<<BB, 256, 0, stream>>>(x, Bpack, b1, W2, b2, Ws1, bs1, Ws2,
                                        bs2, latent, pred);
  decode_kernel<<<BB, 256, 0, stream>>>(Bpack, Tpack, b1, W2, b2, init_set,
                                        latent, recon);
}